// GAT_19344532701548
// MI455X (gfx1250) — compile-verified
//
#include <hip/hip_runtime.h>
#include <math.h>

#define N_NODES 100000
#define N_EDGES 1600000
#define NE2     (N_EDGES + N_NODES)   // edges + self loops
#define SLOPE   0.2f

typedef float v2f __attribute__((ext_vector_type(2)));
typedef float v8f __attribute__((ext_vector_type(8)));

// ---------------------------------------------------------------------------
// Dense transform: y[M,DOUT] = x[M,DIN] @ W[DIN,DOUT] + b
// M = N_NODES (multiple of 16). One wave computes MT 16x16 output tiles with
// V_WMMA_F32_16X16X4_F32, K in steps of 4. B-fragments hoisted across tiles.
//
// fp32 WMMA operand layout (wave32):
//   A (16x4):  lane l<16 holds A[l][0],A[l][1];  lane l>=16 holds A[l-16][2],A[l-16][3]
//   B (4x16):  lane l<16 holds B[0][l],B[1][l];  lane l>=16 holds B[2][l-16],B[3][l-16]
//   C/D (16x16): vgpr r = C[r + 8*(lane>>4)][lane&15]
//
// Guards are compile-time dead for DIN=16/32; the DIN=11 tail uses clamped
// addresses + value selects (v_cndmask) so EXEC stays all-ones (WMMA req).
// ---------------------------------------------------------------------------
template <int DIN, int DOUT, int MT>
__global__ __launch_bounds__(128)
void gemm_wmma_f32(const float* __restrict__ x, const float* __restrict__ W,
                   const float* __restrict__ bias, float* __restrict__ y) {
  constexpr int KSTEPS = (DIN + 3) / 4;
  const int lane = threadIdx.x & 31;
  const int wave = threadIdx.x >> 5;
  const int col  = blockIdx.y * 16 + (lane & 15);   // N for B/C/D
  const int row  = lane & 15;                       // M for A
  const int kb   = (lane >> 4) * 2;                 // K sub-base {0,2}
  const int tileBase = (blockIdx.x * 4 + wave) * MT;

  // B fragments, invariant across M tiles
  v2f bf[KSTEPS];
#pragma unroll
  for (int ks = 0; ks < KSTEPS; ++ks) {
    const int k0 = ks * 4 + kb;
    if (ks * 4 + 3 < DIN) {                 // folds at unroll time
      bf[ks].x = W[k0 * DOUT + col];
      bf[ks].y = W[(k0 + 1) * DOUT + col];
    } else {                                // branchless tail (DIN=11 only)
      const int ka = (k0 < DIN) ? k0 : 0;
      const int kc = (k0 + 1 < DIN) ? k0 + 1 : 0;
      float b0 = W[ka * DOUT + col];
      float b1 = W[kc * DOUT + col];
      bf[ks].x = (k0 < DIN) ? b0 : 0.f;
      bf[ks].y = (k0 + 1 < DIN) ? b1 : 0.f;
    }
  }
  const float bv = bias[col];

#pragma unroll
  for (int t = 0; t < MT; ++t) {
    const int tile = tileBase + t;
    if (tile * 16 >= N_NODES) return;       // wave-uniform scalar branch
    const int node0 = tile * 16;
    const float* xrow = x + (size_t)(node0 + row) * DIN;

    v8f acc = {0.f, 0.f, 0.f, 0.f, 0.f, 0.f, 0.f, 0.f};
#pragma unroll
    for (int ks = 0; ks < KSTEPS; ++ks) {
      const int k0 = ks * 4 + kb;
      v2f a;
      if (ks * 4 + 3 < DIN) {
        a.x = xrow[k0];
        a.y = xrow[k0 + 1];
      } else {
        const int ka = (k0 < DIN) ? k0 : 0;
        const int kc = (k0 + 1 < DIN) ? k0 + 1 : 0;
        float a0 = xrow[ka];
        float a1 = xrow[kc];
        a.x = (k0 < DIN) ? a0 : 0.f;
        a.y = (k0 + 1 < DIN) ? a1 : 0.f;
      }
      acc = __builtin_amdgcn_wmma_f32_16x16x4_f32(false, a, false, bf[ks],
                                                  (short)0, acc, false, false);
    }

    const int mhalf = (lane >> 4) * 8;
#pragma unroll
    for (int r = 0; r < 8; ++r)
      y[(size_t)(node0 + mhalf + r) * DOUT + col] = acc[r] + bv;
  }
}

// ---------------------------------------------------------------------------
__global__ __launch_bounds__(256)
void fill_kernel(float* __restrict__ p, float v, int n) {
  int i = blockIdx.x * blockDim.x + threadIdx.x;
  if (i < n) p[i] = v;
}

// exact float atomic max via sign-aware integer atomics
__device__ __forceinline__ void atomicMaxF(float* a, float v) {
  if (v >= 0.f) atomicMax((int*)a, __float_as_int(v));
  else          atomicMin((unsigned int*)a, __float_as_uint(v));
}

// per-edge logit: e = dot(leaky_relu(xl[src]+xr[dst]), att); segment max into m
__global__ __launch_bounds__(256)
void edge_logit(const int* __restrict__ srcA, const int* __restrict__ dstA,
                const float* __restrict__ xl, const float* __restrict__ xr,
                const float* __restrict__ att, float* __restrict__ e,
                float* __restrict__ m, int dout) {
  __shared__ float satt[64];
  if ((int)threadIdx.x < dout) satt[threadIdx.x] = att[threadIdx.x];
  __syncthreads();
  int i = blockIdx.x * blockDim.x + threadIdx.x;
  if (i >= NE2) return;
  int s, d;
  if (i < N_EDGES) { s = srcA[i]; d = dstA[i]; }
  else             { s = d = i - N_EDGES; }
  const float4* pl = (const float4*)(xl + (size_t)s * dout);
  const float4* pr = (const float4*)(xr + (size_t)d * dout);
  float acc = 0.f;
  const int ch = dout >> 2;
  for (int c = 0; c < ch; ++c) {
    float4 a = pl[c], bb = pr[c];
    float v;
    v = a.x + bb.x; acc += (v > 0.f ? v : SLOPE * v) * satt[4 * c + 0];
    v = a.y + bb.y; acc += (v > 0.f ? v : SLOPE * v) * satt[4 * c + 1];
    v = a.z + bb.z; acc += (v > 0.f ? v : SLOPE * v) * satt[4 * c + 2];
    v = a.w + bb.w; acc += (v > 0.f ? v : SLOPE * v) * satt[4 * c + 3];
  }
  e[i] = acc;
  atomicMaxF(m + d, acc);
}

// p = exp(e - m[dst]) (in place over e); segment sum into denom
__global__ __launch_bounds__(256)
void edge_exp(const int* __restrict__ dstA, const float* __restrict__ m,
              float* __restrict__ e, float* __restrict__ denom) {
  int i = blockIdx.x * blockDim.x + threadIdx.x;
  if (i >= NE2) return;
  int d = (i < N_EDGES) ? dstA[i] : i - N_EDGES;
  float p = __expf(e[i] - m[d]);
  e[i] = p;
  atomicAdd(denom + d, p);
}

// out[dst] += (p/denom[dst]) * xl[src]; chunk-major thread mapping so a wave
// covers contiguous float4 chunks of the same edge row (coalesced gather+atomics)
__global__ __launch_bounds__(256)
void edge_agg(const int* __restrict__ srcA, const int* __restrict__ dstA,
              const float* __restrict__ xl, const float* __restrict__ p,
              const float* __restrict__ denom, float* __restrict__ out,
              int dout, int lgch) {
  unsigned t    = blockIdx.x * blockDim.x + threadIdx.x;
  unsigned edge = t >> lgch;
  unsigned ch   = t & ((1u << lgch) - 1u);
  if (edge >= NE2) return;
  int s, d;
  if (edge < N_EDGES) { s = srcA[edge]; d = dstA[edge]; }
  else                { s = d = (int)edge - N_EDGES; }
  const float alpha = p[edge] / (denom[d] + 1e-16f);
  const float4 a = *(const float4*)(xl + (size_t)s * dout + ch * 4);
  float* o = out + (size_t)d * dout + ch * 4;
  atomicAdd(o + 0, alpha * a.x);
  atomicAdd(o + 1, alpha * a.y);
  atomicAdd(o + 2, alpha * a.z);
  atomicAdd(o + 3, alpha * a.w);
}

// in-place epilogue: h += bias (broadcast over rows), optional ReLU
__global__ __launch_bounds__(256)
void epilogue(float* __restrict__ h, const float* __restrict__ bias,
              int doutMask, int doRelu, int n) {
  int i = blockIdx.x * blockDim.x + threadIdx.x;
  if (i >= n) return;
  float v = h[i] + bias[i & doutMask];
  if (doRelu) v = v > 0.f ? v : 0.f;
  h[i] = v;
}

// ---------------------------------------------------------------------------
extern "C" void kernel_launch(void* const* d_in, const int* in_sizes, int n_in,
                              void* d_out, int out_size, void* d_ws, size_t ws_size,
                              hipStream_t stream) {
  const float* x    = (const float*)d_in[0];
  const int*   ei   = (const int*)d_in[1];
  const int*   srcA = ei;
  const int*   dstA = ei + N_EDGES;

  struct Layer {
    const float *Wl, *bl, *Wr, *br, *att, *bias;
    int dout;
  } L[3];
  static const int douts[3] = {16, 32, 64};
  for (int l = 0; l < 3; ++l) {
    int base = 2 + 6 * l;
    L[l].Wl   = (const float*)d_in[base + 0];
    L[l].bl   = (const float*)d_in[base + 1];
    L[l].Wr   = (const float*)d_in[base + 2];
    L[l].br   = (const float*)d_in[base + 3];
    L[l].att  = (const float*)d_in[base + 4];
    L[l].bias = (const float*)d_in[base + 5];
    L[l].dout = douts[l];
  }

  // workspace layout (floats)
  float* ws = (float*)d_ws;
  size_t off = 0;
  float* hA   = ws + off; off += (size_t)N_NODES * 16;   // layer1 output
  float* hB   = ws + off; off += (size_t)N_NODES * 32;   // layer2 output
  float* xl   = ws + off; off += (size_t)N_NODES * 64;
  float* xr   = ws + off; off += (size_t)N_NODES * 64;
  float* mbuf = ws + off; off += (size_t)N_NODES;
  float* dbuf = ws + off; off += (size_t)N_NODES;
  float* ebuf = ws + off; off += (size_t)NE2;

  const float* layerIn[3]  = {x, hA, hB};
  float*       layerOut[3] = {hA, hB, (float*)d_out};

  constexpr int MT     = 4;                       // M-tiles per wave
  const int tiles      = N_NODES / 16;            // 6250 exactly
  const int gemmBx     = (tiles + 4 * MT - 1) / (4 * MT);
  const int edgeBl     = (NE2 + 255) / 256;
  const int nodeBl     = (N_NODES + 255) / 256;

  for (int l = 0; l < 3; ++l) {
    const int dout = L[l].dout;
    const int lgch = (dout == 16) ? 2 : (dout == 32) ? 3 : 4;
    float* acc = layerOut[l];
    const int accN = N_NODES * dout;

    // 1) dense transforms via fp32 WMMA (templated on shapes)
    dim3 ggrid(gemmBx, dout / 16, 1);
    if (l == 0) {
      gemm_wmma_f32<11, 16, MT><<<ggrid, 128, 0, stream>>>(layerIn[l], L[l].Wl, L[l].bl, xl);
      gemm_wmma_f32<11, 16, MT><<<ggrid, 128, 0, stream>>>(layerIn[l], L[l].Wr, L[l].br, xr);
    } else if (l == 1) {
      gemm_wmma_f32<16, 32, MT><<<ggrid, 128, 0, stream>>>(layerIn[l], L[l].Wl, L[l].bl, xl);
      gemm_wmma_f32<16, 32, MT><<<ggrid, 128, 0, stream>>>(layerIn[l], L[l].Wr, L[l].br, xr);
    } else {
      gemm_wmma_f32<32, 64, MT><<<ggrid, 128, 0, stream>>>(layerIn[l], L[l].Wl, L[l].bl, xl);
      gemm_wmma_f32<32, 64, MT><<<ggrid, 128, 0, stream>>>(layerIn[l], L[l].Wr, L[l].br, xr);
    }

    // 2) init segment buffers + accumulator
    fill_kernel<<<nodeBl, 256, 0, stream>>>(mbuf, -__builtin_inff(), N_NODES);
    fill_kernel<<<nodeBl, 256, 0, stream>>>(dbuf, 0.f, N_NODES);
    fill_kernel<<<(accN + 255) / 256, 256, 0, stream>>>(acc, 0.f, accN);

    // 3) per-edge logits + segment max
    edge_logit<<<edgeBl, 256, 0, stream>>>(srcA, dstA, xl, xr, L[l].att, ebuf, mbuf, dout);

    // 4) exp + segment sum
    edge_exp<<<edgeBl, 256, 0, stream>>>(dstA, mbuf, ebuf, dbuf);

    // 5) alpha-weighted scatter-aggregate
    const unsigned aggThreads = (unsigned)NE2 << lgch;
    edge_agg<<<(aggThreads + 255) / 256, 256, 0, stream>>>(srcA, dstA, xl, ebuf, dbuf,
                                                           acc, dout, lgch);

    // 6) bias + ReLU (no ReLU on final layer)
    epilogue<<<(accN + 255) / 256, 256, 0, stream>>>(acc, L[l].bias, dout - 1,
                                                     (l < 2) ? 1 : 0, accN);
  }
}